// RSS_SS2D_Vmamba_6493990552155
// MI455X (gfx1250) — compile-verified
//
#include <hip/hip_runtime.h>
#include <hip/hip_bf16.h>

// ---------------------------------------------------------------------------
// Problem dims (compile-time constants from the reference)
// ---------------------------------------------------------------------------
#define B_    8
#define C1_   256
#define HH_   32
#define WW_   32
#define L_    1024      // H*W
#define DM_   128
#define DI_   256
#define N_    16
#define R_    8
#define K_    4
#define HID_  256
#define XPC_  (R_ + 2*N_)   // 40 projection channels per direction

typedef __attribute__((ext_vector_type(2))) float v2f;
typedef __attribute__((ext_vector_type(8))) float v8f;

__device__ __forceinline__ float silu_f(float x) { return x / (1.0f + __expf(-x)); }

// A-fragment load: contiguous K (one b64) or strided K (two b32).
template<bool CONTIG>
__device__ __forceinline__ v2f load_a(const float* p, long long strideK)
{
  if constexpr (CONTIG) {
    return *(const v2f*)p;                 // 8B-aligned: k offset is even
  } else {
    v2f v; v[0] = p[0]; v[1] = p[strideK]; return v;
  }
}

// ---------------------------------------------------------------------------
// Generic fp32 WMMA GEMM:  out[m][n] = epilogue( sum_k A[m][k] * W[n][k] )
//   A element addr   = (m/L)*aBatch + (m%L)*aRow + k*aK           (k < kSplit)
//   A2 element addr  = (m/L)*a2Batch + (m%L)*a2Row + (k-kSplit)*a2K (K-concat)
//   W element addr   = n*ldw + k   (weights [out,in], row-major)
//   out element addr = (m/L)*oBatch + (m%L)*oRow + n*oCol
//
// 256 threads = 8 waves / block; each wave owns one 16x16 output tile.
// K stepped by 4 with v_wmma_f32_16x16x4_f32 (exact fp32, RNE).
// 2-stage software pipeline: iteration i+1's A/W loads are issued before
// iteration i's WMMA so the load latency overlaps the matrix op.
//
// VGPR layouts per CDNA5 ISA 7.12.2:
//   A 16x4 : lanes 0-15 hold K={0,1}, lanes 16-31 hold K={2,3}; M = lane%16
//   B 4x16 : same K-half split; N = lane%16
//   C/D    : acc[v] -> row v (lanes<16) or row v+8 (lanes>=16); col = lane%16
//
// No divergence in the K-loop: the A/A2 split is two sequential loops, and the
// out-of-range column guard is a clamped index (garbage stays confined to the
// clamped lane's own column, which is never stored).
// ---------------------------------------------------------------------------
template<bool CONTIG>
__global__ __launch_bounds__(256) void gemm_wmma_f32(
    const float* __restrict__ A,  long long aBatch,  long long aRow,  long long aK,
    const float* __restrict__ A2, long long a2Batch, long long a2Row, long long a2K, int kSplit,
    const float* __restrict__ W,  int ldw,
    int M, int Ncols, int Ktot,
    const float* __restrict__ bias,
    const float* __restrict__ scale, const float* __restrict__ shift,
    const float* __restrict__ resid, long long rBatch, long long rRow,
    int act,
    float* __restrict__ out, long long oBatch, long long oRow, long long oCol)
{
  const int wave = threadIdx.x >> 5;
  const int lane = threadIdx.x & 31;
  const int tm   = (blockIdx.x * 8 + wave) * 16;
  const int tn   = blockIdx.y * 16;
  if (tm >= M) return;                         // wave-uniform

  const int lm = lane & 15;
  const int kh = (lane >> 4) << 1;             // K offset: 0 (lanes 0-15) / 2 (lanes 16-31)

  const int arow     = tm + lm;
  const long long ab = arow / L_;
  const long long al = arow % L_;
  const int  wcol    = tn + lm;
  const bool wok     = (wcol < Ncols);
  const int  wc      = wok ? wcol : (Ncols - 1);   // clamped: loads stay in-bounds

  const float*     pW    = W + (long long)wc * ldw + kh;
  const float*     pA    = A + ab * aBatch + al * aRow + (long long)kh * aK;
  const long long  aStep = 4 * aK;
  const int        kEnd1 = A2 ? kSplit : Ktot;

  v8f acc = {0.f, 0.f, 0.f, 0.f, 0.f, 0.f, 0.f, 0.f};

  // ---- region 1: A ----
  {
    v2f av = load_a<CONTIG>(pA, aK);
    v2f bv = *(const v2f*)pW;
    for (int k0 = 4; k0 < kEnd1; k0 += 4) {
      pA += aStep;
      pW += 4;
      const v2f avn = load_a<CONTIG>(pA, aK);
      const v2f bvn = *(const v2f*)pW;
      acc = __builtin_amdgcn_wmma_f32_16x16x4_f32(false, av, false, bv,
                                                  (short)0, acc, false, false);
      av = avn; bv = bvn;
    }
    acc = __builtin_amdgcn_wmma_f32_16x16x4_f32(false, av, false, bv,
                                                (short)0, acc, false, false);
  }

  // ---- region 2: A2 (K-concat tail) ----
  if (A2) {
    const float*    pA2    = A2 + ab * a2Batch + al * a2Row + (long long)kh * a2K;
    const float*    pW2    = W + (long long)wc * ldw + kh + kSplit;
    const long long a2Step = 4 * a2K;

    v2f av = load_a<CONTIG>(pA2, a2K);
    v2f bv = *(const v2f*)pW2;
    for (int k0 = kSplit + 4; k0 < Ktot; k0 += 4) {
      pA2 += a2Step;
      pW2 += 4;
      const v2f avn = load_a<CONTIG>(pA2, a2K);
      const v2f bvn = *(const v2f*)pW2;
      acc = __builtin_amdgcn_wmma_f32_16x16x4_f32(false, av, false, bv,
                                                  (short)0, acc, false, false);
      av = avn; bv = bvn;
    }
    acc = __builtin_amdgcn_wmma_f32_16x16x4_f32(false, av, false, bv,
                                                (short)0, acc, false, false);
  }

  if (!wok) return;
#pragma unroll
  for (int v = 0; v < 8; ++v) {
    const int row = tm + ((lane < 16) ? v : v + 8);
    float val = acc[v];
    if (scale) val = val * scale[wcol] + shift[wcol];   // folded-BN affine
    if (bias)  val += bias[wcol];
    const long long b = row / L_, l = row % L_;
    if (resid) val += resid[b * rBatch + l * rRow + wcol];
    if (act)   val = silu_f(val);
    out[b * oBatch + l * oRow + (long long)wcol * oCol] = val;
  }
}

// ---------------------------------------------------------------------------
// Depthwise 3x3 conv (SAME) + bias + SiLU.
//  in : xz tokens (B,L,2*DI) -> channel d of the first DI
//  out: xc channel-major (B,DI,L)
// ---------------------------------------------------------------------------
__global__ __launch_bounds__(256) void dwconv_kernel(
    const float* __restrict__ xz, const float* __restrict__ cv_w,
    const float* __restrict__ cv_b, float* __restrict__ xc)
{
  const long long idx = (long long)blockIdx.x * blockDim.x + threadIdx.x;
  if (idx >= (long long)B_ * DI_ * L_) return;
  const int l = (int)(idx % L_);
  const int d = (int)((idx / L_) % DI_);
  const int b = (int)(idx / ((long long)L_ * DI_));
  const int hh = l / WW_, ww = l % WW_;
  float acc = cv_b[d];
#pragma unroll
  for (int ky = 0; ky < 3; ++ky) {
    const int y = hh + ky - 1;
    if (y < 0 || y >= HH_) continue;
#pragma unroll
    for (int kx = 0; kx < 3; ++kx) {
      const int x = ww + kx - 1;
      if (x < 0 || x >= WW_) continue;
      acc += xz[((long long)b * L_ + y * WW_ + x) * (2 * DI_) + d] * cv_w[d * 9 + ky * 3 + kx];
    }
  }
  xc[((long long)b * DI_ + d) * L_ + l] = silu_f(acc);
}

// sequence position t of direction k -> spatial index l (bijection on [0,L))
__device__ __forceinline__ int dir_map(int k, int t)
{
  switch (k) {
    case 0:  return t;                                                   // hw
    case 1:  { const int h = t % HH_, w = t / HH_; return h * WW_ + w; } // wh
    case 2:  return L_ - 1 - t;                                          // hw reversed
    default: { const int tt = L_ - 1 - t;
               const int h = tt % HH_, w = tt / HH_; return h * WW_ + w; } // wh reversed
  }
}

// ---------------------------------------------------------------------------
// Selective scan.  One block per (b,k); thread d owns channel d's 16-state.
// Folds the dt projection (R=8 -> DI) + softplus into the time loop; the
// shared 40-value row (dts,B,C) is broadcast through LDS each step.
// Output written in SPATIAL order so all 4 directions can be summed directly.
// ---------------------------------------------------------------------------
__global__ __launch_bounds__(256) void scan_kernel(
    const float* __restrict__ xc, const float* __restrict__ P,
    const float* __restrict__ dt_w, const float* __restrict__ dt_b,
    const float* __restrict__ A_log, const float* __restrict__ Dskip,
    float* __restrict__ ys)
{
  const int bk = blockIdx.x;
  const int b = bk / K_, k = bk % K_;
  const int d = threadIdx.x;

  float dtw[R_];
#pragma unroll
  for (int r = 0; r < R_; ++r) dtw[r] = dt_w[((long long)k * DI_ + d) * R_ + r];
  const float dtb = dt_b[k * DI_ + d];
  float Ac[N_];
#pragma unroll
  for (int n = 0; n < N_; ++n) Ac[n] = -__expf(A_log[((long long)k * DI_ + d) * N_ + n]);
  const float dsk = Dskip[k * DI_ + d];

  float h[N_];
#pragma unroll
  for (int n = 0; n < N_; ++n) h[n] = 0.f;

  __shared__ float row[XPC_];
  const float* Pb = P + (((long long)b * K_ + k) * L_) * XPC_;
  const float* xb = xc + ((long long)b * DI_ + d) * L_;
  float*       yb = ys + (((long long)b * K_ + k) * DI_ + d) * L_;

  for (int t = 0; t < L_; ++t) {
    const int l = dir_map(k, t);
    __syncthreads();
    if (d < XPC_) row[d] = Pb[(long long)l * XPC_ + d];
    __syncthreads();

    const float u = xb[l];
    float dtr = dtb;
#pragma unroll
    for (int r = 0; r < R_; ++r) dtr += row[r] * dtw[r];
    const float delta = (dtr > 20.f) ? dtr : log1pf(__expf(dtr));   // softplus
    const float du = delta * u;
    float y = 0.f;
#pragma unroll
    for (int n = 0; n < N_; ++n) {
      h[n] = h[n] * __expf(delta * Ac[n]) + du * row[R_ + n];
      y += h[n] * row[R_ + N_ + n];
    }
    yb[l] = y + dsk * u;
  }
}

// ---------------------------------------------------------------------------
// Sum 4 directions + LayerNorm(DI) + SiLU(z) gate.  One block per (b,l).
// ---------------------------------------------------------------------------
__global__ __launch_bounds__(256) void merge_ln_kernel(
    const float* __restrict__ ys, const float* __restrict__ xz,
    const float* __restrict__ ln_g, const float* __restrict__ ln_b,
    float* __restrict__ yn)
{
  const int bl = blockIdx.x;
  const int d = threadIdx.x;
  const long long b = bl / L_, l = bl % L_;

  float s = 0.f;
#pragma unroll
  for (int k = 0; k < K_; ++k)
    s += ys[(((long long)b * K_ + k) * DI_ + d) * L_ + l];

  __shared__ float red[DI_];
  red[d] = s; __syncthreads();
  for (int off = DI_ / 2; off > 0; off >>= 1) {
    if (d < off) red[d] += red[d + off];
    __syncthreads();
  }
  const float mu = red[0] * (1.0f / DI_);
  __syncthreads();
  const float dv = s - mu;
  red[d] = dv * dv; __syncthreads();
  for (int off = DI_ / 2; off > 0; off >>= 1) {
    if (d < off) red[d] += red[d + off];
    __syncthreads();
  }
  const float var = red[0] * (1.0f / DI_);

  const float v = dv * rsqrtf(var + 1e-5f) * ln_g[d] + ln_b[d];
  const float z = xz[(long long)bl * (2 * DI_) + DI_ + d];
  yn[(long long)bl * DI_ + d] = v * silu_f(z);
}

// ---------------------------------------------------------------------------
// Host side
// ---------------------------------------------------------------------------
// Workspace layout (float offsets); total 22,282,240 floats (~85 MB)
#define OFF_TOK0 0LL            // y1 tokens (B,L,DM)
#define OFF_TOK  1048576LL      // working tokens (B,L,DM)
#define OFF_XZ   2097152LL      // in-proj output (B,L,2*DI)
#define OFF_XC   6291456LL      // conv'd xin, channel-major (B,DI,L)
#define OFF_P    8388608LL      // x_proj (B,K,L,40), spatial order
#define OFF_YS   9699328LL      // scan outputs (B,K,DI,L), spatial order
#define OFF_YN   18087936LL     // gated/normed y (B,L,DI)
#define OFF_HID  20185088LL     // MLP hidden (B,L,HID)

struct BlockParams {
  const float *in_w, *in_b, *cv_w, *cv_b, *xp_w, *dt_w, *dt_b;
  const float *A_log, *Dskip, *ln_g, *ln_b, *out_w, *out_b;
  const float *m1_w, *m1_s, *m1_sh, *m2_w, *m2_s, *m2_sh;
};

static inline const float* FP(void* const* d_in, int i) { return (const float*)d_in[i]; }

// Input flattening assumption: setup_inputs() insertion order, depth-first:
//  0: x
//  1-3: cv1 {w, scale, shift}
//  4-22: block0 {attn(13): in_w,in_b,cv_w,cv_b,xp_w,dt_w,dt_b,A_log,Dskip,ln_g,ln_b,out_w,out_b;
//               mlp1(3): w,scale,shift; mlp2(3): w,scale,shift}
//  23-41: block1 (same layout)
//  42-44: cv2 {w, scale, shift}
static BlockParams get_block(void* const* d_in, int bi)
{
  const int base = 4 + bi * 19;
  BlockParams p;
  p.in_w  = FP(d_in, base + 0);  p.in_b  = FP(d_in, base + 1);
  p.cv_w  = FP(d_in, base + 2);  p.cv_b  = FP(d_in, base + 3);
  p.xp_w  = FP(d_in, base + 4);  p.dt_w  = FP(d_in, base + 5);
  p.dt_b  = FP(d_in, base + 6);  p.A_log = FP(d_in, base + 7);
  p.Dskip = FP(d_in, base + 8);  p.ln_g  = FP(d_in, base + 9);
  p.ln_b  = FP(d_in, base + 10); p.out_w = FP(d_in, base + 11);
  p.out_b = FP(d_in, base + 12);
  p.m1_w  = FP(d_in, base + 13); p.m1_s  = FP(d_in, base + 14); p.m1_sh = FP(d_in, base + 15);
  p.m2_w  = FP(d_in, base + 16); p.m2_s  = FP(d_in, base + 17); p.m2_sh = FP(d_in, base + 18);
  return p;
}

extern "C" void kernel_launch(void* const* d_in, const int* in_sizes, int n_in,
                              void* d_out, int out_size, void* d_ws, size_t ws_size,
                              hipStream_t stream)
{
  (void)in_sizes; (void)n_in; (void)out_size; (void)ws_size;

  const float* x      = FP(d_in, 0);
  const float* cv1_w  = FP(d_in, 1);
  const float* cv1_s  = FP(d_in, 2);
  const float* cv1_sh = FP(d_in, 3);
  const float* cv2_w  = FP(d_in, 42);
  const float* cv2_s  = FP(d_in, 43);
  const float* cv2_sh = FP(d_in, 44);

  float* ws   = (float*)d_ws;
  float* tok0 = ws + OFF_TOK0;
  float* tok  = ws + OFF_TOK;
  float* xz   = ws + OFF_XZ;
  float* xc   = ws + OFF_XC;
  float* Pbuf = ws + OFF_P;
  float* ysb  = ws + OFF_YS;
  float* yn   = ws + OFF_YN;
  float* hid  = ws + OFF_HID;

  const dim3 wblk(256);
  const int M = B_ * L_;

  auto gemm = [&](bool contig,
                  const float* A, long long aB, long long aR, long long aK,
                  const float* A2, long long a2B, long long a2R, long long a2K, int kSplit,
                  const float* W, int ldw, int Ncols, int Ktot,
                  const float* bias, const float* scale, const float* shift,
                  const float* resid, long long rB, long long rR, int act,
                  float* out, long long oB, long long oR, long long oC) {
    dim3 grid((M / 16 + 7) / 8, (Ncols + 15) / 16);
    if (contig)
      gemm_wmma_f32<true><<<grid, wblk, 0, stream>>>(A, aB, aR, aK, A2, a2B, a2R, a2K, kSplit,
                                                     W, ldw, M, Ncols, Ktot,
                                                     bias, scale, shift, resid, rB, rR, act,
                                                     out, oB, oR, oC);
    else
      gemm_wmma_f32<false><<<grid, wblk, 0, stream>>>(A, aB, aR, aK, A2, a2B, a2R, a2K, kSplit,
                                                      W, ldw, M, Ncols, Ktot,
                                                      bias, scale, shift, resid, rB, rR, act,
                                                      out, oB, oR, oC);
  };

  // --- cv1: y1 = silu(BN(x @ W1^T)), NCHW input -> token-major output --------
  gemm(false,
       x, (long long)C1_ * L_, 1, L_, nullptr, 0, 0, 0, 0,
       cv1_w, C1_, DM_, C1_,
       nullptr, cv1_s, cv1_sh, nullptr, 0, 0, 1,
       tok0, (long long)L_ * DM_, DM_, 1);

  hipMemcpyAsync(tok, tok0, sizeof(float) * (size_t)B_ * L_ * DM_,
                 hipMemcpyDeviceToDevice, stream);

  for (int bi = 0; bi < 2; ++bi) {
    BlockParams p = get_block(d_in, bi);

    // in_proj: xz = tok @ in_w^T + in_b   (B,L,512)
    gemm(true,
         tok, (long long)L_ * DM_, DM_, 1, nullptr, 0, 0, 0, 0,
         p.in_w, DM_, 2 * DI_, DM_,
         p.in_b, nullptr, nullptr, nullptr, 0, 0, 0,
         xz, (long long)L_ * 2 * DI_, 2 * DI_, 1);

    // depthwise 3x3 + bias + silu -> xc (B,DI,L)
    {
      const long long n = (long long)B_ * DI_ * L_;
      dwconv_kernel<<<(unsigned)((n + 255) / 256), 256, 0, stream>>>(xz, p.cv_w, p.cv_b, xc);
    }

    // x_proj per direction: P[b,kd,l,c] = sum_d xc[b,d,l] * xp_w[kd,c,d]
    for (int kd = 0; kd < K_; ++kd)
      gemm(false,
           xc, (long long)DI_ * L_, 1, L_, nullptr, 0, 0, 0, 0,
           p.xp_w + (long long)kd * XPC_ * DI_, DI_, XPC_, DI_,
           nullptr, nullptr, nullptr, nullptr, 0, 0, 0,
           Pbuf + (long long)kd * L_ * XPC_, (long long)K_ * L_ * XPC_, XPC_, 1);

    // selective scan (dt-proj + softplus folded in)
    scan_kernel<<<B_ * K_, 256, 0, stream>>>(xc, Pbuf, p.dt_w, p.dt_b, p.A_log, p.Dskip, ysb);

    // merge 4 directions + LayerNorm + silu(z) gate
    merge_ln_kernel<<<B_ * L_, 256, 0, stream>>>(ysb, xz, p.ln_g, p.ln_b, yn);

    // out_proj + residual (in place on tok)
    gemm(true,
         yn, (long long)L_ * DI_, DI_, 1, nullptr, 0, 0, 0, 0,
         p.out_w, DI_, DM_, DI_,
         p.out_b, nullptr, nullptr, tok, (long long)L_ * DM_, DM_, 0,
         tok, (long long)L_ * DM_, DM_, 1);

    // MLP: hid = silu(BN(tok @ m1^T))
    gemm(true,
         tok, (long long)L_ * DM_, DM_, 1, nullptr, 0, 0, 0, 0,
         p.m1_w, DM_, HID_, DM_,
         nullptr, p.m1_s, p.m1_sh, nullptr, 0, 0, 1,
         hid, (long long)L_ * HID_, HID_, 1);

    // tok = tok + BN(hid @ m2^T)
    gemm(true,
         hid, (long long)L_ * HID_, HID_, 1, nullptr, 0, 0, 0, 0,
         p.m2_w, HID_, DM_, HID_,
         nullptr, p.m2_s, p.m2_sh, tok, (long long)L_ * DM_, DM_, 0,
         tok, (long long)L_ * DM_, DM_, 1);
  }

  // --- cv2 on concat([y1, y2], C): K split across tok0 / tok; NCHW output ---
  gemm(true,
       tok0, (long long)L_ * DM_, DM_, 1,
       tok,  (long long)L_ * DM_, DM_, 1, DM_,
       cv2_w, 2 * DM_, C1_, 2 * DM_,
       nullptr, cv2_s, cv2_sh, nullptr, 0, 0, 1,
       (float*)d_out, (long long)C1_ * L_, 1, L_);
}